// ACSF_51737176048138
// MI455X (gfx1250) — compile-verified
//
#include <hip/hip_runtime.h>
#include <stdint.h>

#define CUTOFF_F   6.0f
#define PI_F       3.14159265358979323846f
#define OUT_W      70

typedef int   v2i __attribute__((ext_vector_type(2)));
typedef float v2f __attribute__((ext_vector_type(2)));

// cosine cutoff: 0.5*cos(pi*r/6)+0.5  (native v_cos_f32 path)
__device__ __forceinline__ float fc_cut(float r) {
    return fmaf(0.5f, __cosf((PI_F / CUTOFF_F) * r), 0.5f);
}

// Stage n floats (n <= blockDim.x) from global into LDS using the CDNA5
// async Global->LDS path (ASYNCcnt). Caller must __syncthreads() after.
__device__ __forceinline__ void stage_to_lds_async(float* lds_dst,
                                                   const float* __restrict__ g_src,
                                                   int n, int tid) {
#if defined(__gfx1250__)
    if (tid < n) {
        // low 32 bits of a flat pointer into the LDS aperture = wave-relative LDS byte offset
        unsigned lds_off = (unsigned)(uintptr_t)(lds_dst + tid);
        const float* gp = g_src + tid;
        asm volatile("global_load_async_to_lds_b32 %0, %1, off"
                     :: "v"(lds_off), "v"(gp) : "memory");
    }
    asm volatile("s_wait_asynccnt 0" ::: "memory");
#else
    if (tid < n) lds_dst[tid] = g_src[tid];
#endif
}

__global__ void acsf_zero(float* __restrict__ out, int n) {
    int i = blockIdx.x * blockDim.x + threadIdx.x;
    if (i < n) out[i] = 0.0f;
}

// ---- G2: one thread per edge, 8 atomics into out[tgt, 2*i + an_src] ----
__global__ void acsf_g2(const int* __restrict__ an,
                        const int* __restrict__ e_src,
                        const int* __restrict__ e_tgt,
                        const float* __restrict__ D,
                        const float* __restrict__ g2_etas,   // [4,8]
                        float* __restrict__ out, int E) {
    __shared__ float s_eta[32];
    stage_to_lds_async(s_eta, g2_etas, 32, threadIdx.x);
    __syncthreads();

    int stride = gridDim.x * blockDim.x;
    for (int e = blockIdx.x * blockDim.x + threadIdx.x; e < E; e += stride) {
        int   s = __builtin_nontemporal_load(e_src + e);   // streamed once -> TH_NT
        int   t = __builtin_nontemporal_load(e_tgt + e);
        float d = __builtin_nontemporal_load(D + e);
        int as = an[s];                                    // gathered, keep cached
        int at = an[t];
        float cut2 = fc_cut(d);          // reference applies no range mask in G2
        float d2   = d * d;
        const float* eta = &s_eta[(as * 2 + at) * 8];
        float* row = out + (size_t)t * OUT_W + as;
#pragma unroll
        for (int i = 0; i < 8; ++i) {
            float v = cut2 * __expf(-eta[i] * d2);
            atomicAdd(&row[2 * i], v);   // non-returning global_atomic_add_f32 (L2-resident)
        }
    }
}

// ---- G4 per-triplet body: 18 atomics into out[seg, 16 + ((i*2+j)*3+l)*3 + trip] ----
__device__ __forceinline__ void g4_triplet(int ba, int ca, float c,
                                           const int* __restrict__ an,
                                           const int* __restrict__ e_src,
                                           const int* __restrict__ e_tgt,
                                           const float* __restrict__ D,
                                           const float* __restrict__ s_eta,
                                           const float* __restrict__ s_zeta,
                                           const float* __restrict__ s_lmda,
                                           float* __restrict__ out) {
    if (ba <= ca) return;                               // m3 dedup mask (halves the work)

    float Rba = D[ba];
    float Rca = D[ca];
    float Rbc2 = fmaf(Rba, Rba, fmaf(Rca, Rca, -2.0f * Rba * Rca * c));
    float Rbc  = sqrtf(fmaxf(Rbc2, 1e-12f));
    if (!((Rbc < CUTOFF_F) && (Rba < CUTOFF_F) && (Rca < CUTOFF_F)))
        return;                                         // cut3 == 0 -> nothing to add

    // species gathers only for surviving triplets
    int sb  = an[e_src[ba]];
    int seg = e_tgt[ba];
    int tb  = an[seg];
    int sc  = an[e_src[ca]];

    float cut3 = fc_cut(Rca) * fc_cut(Rbc) * fc_cut(Rba);
    float r2   = fmaf(Rbc, Rbc, fmaf(Rba, Rba, Rca * Rca));

    int trip = sb + sc;                                 // (0,0)->0 (0,1)/(1,0)->1 (1,1)->2
    int k    = tb * 3 + trip;                           // parameter row

    // cl[j][l] = 2^(1-z) * |1 + lmda*c|^z  ==  exp2(z*log2(b) + (1-z))
    float clz[2][3];
#pragma unroll
    for (int j = 0; j < 2; ++j) {
        float base = fabsf(fmaf(s_lmda[k * 2 + j], c, 1.0f));
        float lb   = __log2f(base);
#pragma unroll
        for (int l = 0; l < 3; ++l) {
            float z = s_zeta[k * 3 + l];
            clz[j][l] = exp2f(fmaf(z, lb, 1.0f - z));
        }
    }

    float* row = out + (size_t)seg * OUT_W + 16 + trip;
#pragma unroll
    for (int i = 0; i < 3; ++i) {
        float rad = cut3 * __expf(-s_eta[k * 3 + i] * r2);
#pragma unroll
        for (int j = 0; j < 2; ++j) {
#pragma unroll
            for (int l = 0; l < 3; ++l) {
                atomicAdd(&row[(((i * 2 + j) * 3 + l) * 3)], rad * clz[j][l]);
            }
        }
    }
}

// ---- G4: two triplets per lane; b64 NT loads on the 24 MB index/cos streams ----
__global__ void acsf_g4(const int* __restrict__ an,
                        const int* __restrict__ e_src,
                        const int* __restrict__ e_tgt,
                        const float* __restrict__ D,
                        const int* __restrict__ id_ba,
                        const int* __restrict__ id_ca,
                        const float* __restrict__ cosphi,
                        const float* __restrict__ g4_etas,   // [6,3]
                        const float* __restrict__ g4_zetas,  // [6,3]
                        const float* __restrict__ g4_lmdas,  // [6,2]
                        float* __restrict__ out, int T) {
    __shared__ float s_eta[18];
    __shared__ float s_zeta[18];
    __shared__ float s_lmda[12];
    stage_to_lds_async(s_eta,  g4_etas,  18, threadIdx.x);
    stage_to_lds_async(s_zeta, g4_zetas, 18, threadIdx.x);
    stage_to_lds_async(s_lmda, g4_lmdas, 12, threadIdx.x);
    __syncthreads();

    int stride2 = gridDim.x * blockDim.x * 2;
    for (int t0 = (blockIdx.x * blockDim.x + threadIdx.x) * 2; t0 < T; t0 += stride2) {
        if (t0 + 1 < T) {
            v2i ba2 = __builtin_nontemporal_load((const v2i*)(id_ba  + t0));
            v2i ca2 = __builtin_nontemporal_load((const v2i*)(id_ca  + t0));
            v2f c2  = __builtin_nontemporal_load((const v2f*)(cosphi + t0));
            g4_triplet(ba2.x, ca2.x, c2.x, an, e_src, e_tgt, D, s_eta, s_zeta, s_lmda, out);
            g4_triplet(ba2.y, ca2.y, c2.y, an, e_src, e_tgt, D, s_eta, s_zeta, s_lmda, out);
        } else {
            int   ba = __builtin_nontemporal_load(id_ba  + t0);
            int   ca = __builtin_nontemporal_load(id_ca  + t0);
            float c  = __builtin_nontemporal_load(cosphi + t0);
            g4_triplet(ba, ca, c, an, e_src, e_tgt, D, s_eta, s_zeta, s_lmda, out);
        }
    }
}

extern "C" void kernel_launch(void* const* d_in, const int* in_sizes, int n_in,
                              void* d_out, int out_size, void* d_ws, size_t ws_size,
                              hipStream_t stream) {
    const int*   an     = (const int*)  d_in[0];   // [N]
    const int*   eidx   = (const int*)  d_in[1];   // [2,E] row-major
    const float* D      = (const float*)d_in[2];   // [E]
    const int*   id_ba  = (const int*)  d_in[3];   // [T]
    const int*   id_ca  = (const int*)  d_in[4];   // [T]
    const float* cosphi = (const float*)d_in[5];   // [T]
    const float* g2e    = (const float*)d_in[6];   // [4,8]
    const float* g4e    = (const float*)d_in[7];   // [6,3]
    const float* g4z    = (const float*)d_in[8];   // [6,3]
    const float* g4l    = (const float*)d_in[9];   // [6,2]
    float* out = (float*)d_out;

    int E = in_sizes[2];
    int T = in_sizes[3];
    const int* e_src = eidx;
    const int* e_tgt = eidx + E;

    const int BZ = 256; // 8 wave32 waves per block
    acsf_zero<<<(out_size + BZ - 1) / BZ, BZ, 0, stream>>>(out, out_size);
    acsf_g2 <<<(E + BZ - 1) / BZ, BZ, 0, stream>>>(an, e_src, e_tgt, D, g2e, out, E);
    int tthreads = (T + 1) / 2;
    acsf_g4 <<<(tthreads + BZ - 1) / BZ, BZ, 0, stream>>>(an, e_src, e_tgt, D, id_ba, id_ca,
                                                          cosphi, g4e, g4z, g4l, out, T);
}